// LabelizerNet_5557687681080
// MI455X (gfx1250) — compile-verified
//
#include <hip/hip_runtime.h>

#define THREADS 256
#define C_CH 3
#define HIN 84
#define WIN 84
#define HP 40
#define WP 40
#define CH_ELEMS (HIN * WIN)        // 7056 floats per channel
#define IMG_ELEMS (C_CH * CH_ELEMS) // 21168 floats per image
#define CH_VEC (CH_ELEMS / 4)       // 1764 float4 per channel

__global__ __launch_bounds__(THREADS)
void labelizer_kernel(const float* __restrict__ states, float* __restrict__ out)
{
    __shared__ __align__(16) float s_img[CH_ELEMS];   // staged channel (async-LDS target)
    __shared__ float s_hsum[HIN * WP];                // horizontal 6-tap sums
    __shared__ float s_pool[C_CH * HP * WP];          // pooled RGB
    __shared__ int s_pmax[3];
    __shared__ int s_pmin[3];

    const int tid = threadIdx.x;
    const int img = blockIdx.x;
    const float* gimg = states + (size_t)img * IMG_ELEMS;

    if (tid < 3) { s_pmax[tid] = -1; s_pmin[tid] = 1; }

    for (int c = 0; c < C_CH; ++c) {
        // ---- CDNA5 async global -> LDS copy of one channel (ASYNCcnt path) ----
        {
            const float4* gsrc = (const float4*)(gimg + c * CH_ELEMS);
            for (int i = tid; i < CH_VEC; i += THREADS) {
                // low 32 bits of a flat shared pointer == LDS byte offset
                unsigned lofs = (unsigned)(unsigned long long)(const void*)(s_img + i * 4);
                unsigned long long gaddr = (unsigned long long)(const void*)(gsrc + i);
                asm volatile("global_load_async_to_lds_b128 %0, %1, off"
                             :: "v"(lofs), "v"(gaddr) : "memory");
            }
            asm volatile("s_wait_asynccnt 0" ::: "memory");
        }
        __syncthreads();

        // ---- horizontal 6-tap, stride 2: 84 rows x 40 cols ----
        for (int i = tid; i < HIN * WP; i += THREADS) {
            int y = i / WP, x = i - y * WP;
            const float* row = s_img + y * WIN + 2 * x;
            s_hsum[i] = ((row[0] + row[1]) + (row[2] + row[3])) + (row[4] + row[5]);
        }
        __syncthreads();

        // ---- vertical 6-tap, stride 2, then /36 ----
        for (int i = tid; i < HP * WP; i += THREADS) {
            int y = i / WP, x = i - y * WP;
            const float* col = s_hsum + 2 * y * WP + x;
            float sv = ((col[0] + col[WP]) + (col[2 * WP] + col[3 * WP]))
                       + (col[4 * WP] + col[5 * WP]);
            s_pool[c * (HP * WP) + i] = sv / 36.0f;
        }
        __syncthreads(); // also guards s_img/s_hsum reuse for the next channel
    }

    // ---- HSV + threshold mask + per-panel running min/max in registers ----
    int lmax0 = -2, lmax1 = -2, lmax2 = -2;   // -2 sentinel: never raises pmax (floor is -1)
    int lmin0 =  2, lmin1 =  2, lmin2 =  2;   //  2 sentinel: never lowers pmin (ceil is 1)

    for (int i = tid; i < HP * WP; i += THREADS) {
        float r = s_pool[i];
        float g = s_pool[HP * WP + i];
        float b = s_pool[2 * HP * WP + i];

        float maxc = fmaxf(r, fmaxf(g, b));
        float minc = fminf(r, fminf(g, b));
        float delta = maxc - minc;
        float s = (maxc == 0.0f) ? 0.0f : delta / maxc;
        float sd = (delta == 0.0f) ? 1.0f : delta;
        float rc = (maxc - r) / sd;
        float gc = (maxc - g) / sd;
        float bc = (maxc - b) / sd;
        float h6 = (r == maxc) ? (bc - gc)
                 : ((g == maxc) ? (2.0f + rc - bc) : (4.0f + gc - rc));
        float h = h6 / 6.0f;
        h -= floorf(h);                 // python-style mod 1.0 (wraps negatives)
        if (delta == 0.0f) h = 0.0f;

        bool satval = (s > 0.5f) && (maxc > 0.5f);
        int m = 0;
        if (satval) {
            if (h >= 0.1f && h <= 0.2f)       m = 1;
            else if (h >= 0.55f && h <= 0.7f) m = -1;
        }

        int x = i - (i / WP) * WP;
        if (x < 39) {                   // column 39 excluded by panel bounds
            int p = x / 13;
            lmax0 = (p == 0 && m > lmax0) ? m : lmax0;
            lmax1 = (p == 1 && m > lmax1) ? m : lmax1;
            lmax2 = (p == 2 && m > lmax2) ? m : lmax2;
            lmin0 = (p == 0 && m < lmin0) ? m : lmin0;
            lmin1 = (p == 1 && m < lmin1) ? m : lmin1;
            lmin2 = (p == 2 && m < lmin2) ? m : lmin2;
        }
    }

    // ---- wave32 tree reduction, then 6 LDS atomics per wave ----
    #pragma unroll
    for (int off = 16; off > 0; off >>= 1) {
        lmax0 = max(lmax0, __shfl_xor(lmax0, off, 32));
        lmax1 = max(lmax1, __shfl_xor(lmax1, off, 32));
        lmax2 = max(lmax2, __shfl_xor(lmax2, off, 32));
        lmin0 = min(lmin0, __shfl_xor(lmin0, off, 32));
        lmin1 = min(lmin1, __shfl_xor(lmin1, off, 32));
        lmin2 = min(lmin2, __shfl_xor(lmin2, off, 32));
    }
    if ((tid & 31) == 0) {
        atomicMax(&s_pmax[0], lmax0);
        atomicMax(&s_pmax[1], lmax1);
        atomicMax(&s_pmax[2], lmax2);
        atomicMin(&s_pmin[0], lmin0);
        atomicMin(&s_pmin[1], lmin1);
        atomicMin(&s_pmin[2], lmin2);
    }
    __syncthreads();

    if (tid < 3)      out[img * 6 + tid] = (float)s_pmax[tid];
    else if (tid < 6) out[img * 6 + tid] = (float)(-s_pmin[tid - 3]);
}

extern "C" void kernel_launch(void* const* d_in, const int* in_sizes, int n_in,
                              void* d_out, int out_size, void* d_ws, size_t ws_size,
                              hipStream_t stream)
{
    (void)n_in; (void)d_ws; (void)ws_size; (void)out_size;
    const float* states = (const float*)d_in[0];
    float* out = (float*)d_out;
    int batch = in_sizes[0] / IMG_ELEMS;   // 4096
    labelizer_kernel<<<batch, THREADS, 0, stream>>>(states, out);
}